// RNN_25580825215667
// MI455X (gfx1250) — compile-verified
//
#include <hip/hip_runtime.h>

// 2-layer tanh RNN (H=32, B=512, T=2048) on CDNA5 using fp32 WMMA 16x16x4.
// One wave (32 threads) per 16-batch tile; weights + both hidden states live
// in registers; D-layout -> A-layout transposes go through LDS each step.

typedef float v2f __attribute__((ext_vector_type(2)));
typedef float v8f __attribute__((ext_vector_type(8)));

#define HID 32
#define BB  512
#define TT  2048

__device__ __forceinline__ v8f wmma_f32(v2f a, v2f b, v8f c) {
  // V_WMMA_F32_16X16X4_F32: A 16x4 f32 (2 VGPRs), B 4x16 f32 (2 VGPRs), C/D 16x16 f32 (8 VGPRs)
  return __builtin_amdgcn_wmma_f32_16x16x4_f32(false, a, false, b, (short)0, c, false, false);
}

__device__ __forceinline__ float fast_tanh(float x) {
  // tanh(x) = 1 - 2/(exp(2x)+1); exp(2x) = exp2(x * 2*log2(e))
  float e = __builtin_amdgcn_exp2f(x * 2.885390081777927f);
  return 1.0f - 2.0f * __builtin_amdgcn_rcpf(e + 1.0f);
}

__launch_bounds__(32)
__global__ void rnn2_wmma_kernel(const float* __restrict__ xg,    // [B,T,1]
                                 const float* __restrict__ hid,   // [2,B,H]
                                 const float* __restrict__ Wih0,  // [H,1]
                                 const float* __restrict__ Whh0,  // [H,H]
                                 const float* __restrict__ bih0,  // [H]
                                 const float* __restrict__ bhh0,  // [H]
                                 const float* __restrict__ Wih1,  // [H,H]
                                 const float* __restrict__ Whh1,  // [H,H]
                                 const float* __restrict__ bih1,  // [H]
                                 const float* __restrict__ bhh1,  // [H]
                                 const float* __restrict__ Wfc,   // [1,H]
                                 const float* __restrict__ bfc,   // [1]
                                 float* __restrict__ out)         // [B*T] ++ [2,B,H]
{
  __shared__ __align__(16) float hst[16][34];  // 16x32 tile, stride 34 (8B aligned, bank-spread)
  __shared__ __align__(16) float xbuf[2][16];  // double-buffered x staging

  const int lane = threadIdx.x;   // 0..31
  const int n    = lane & 15;     // N index within a 16-wide tile / A-layout row (M)
  const int hi   = lane >> 4;     // which K-pair half within a K=4 chunk
  const int b0   = blockIdx.x * 16;

  // ---- Preload weights (B-layout: lane holds W^T[k0..k0+1][n + 16*tile], k0 = 4c + 2*hi) ----
  v2f wh0[8][2], wi1[8][2], wh1[8][2], wf[8];
  #pragma unroll
  for (int c = 0; c < 8; ++c) {
    const int k0 = 4 * c + 2 * hi;
    wh0[c][0] = *(const v2f*)(Whh0 + (n     ) * HID + k0);   // W^T[k][j] = W[j][k]
    wh0[c][1] = *(const v2f*)(Whh0 + (n + 16) * HID + k0);
    wi1[c][0] = *(const v2f*)(Wih1 + (n     ) * HID + k0);
    wi1[c][1] = *(const v2f*)(Wih1 + (n + 16) * HID + k0);
    wh1[c][0] = *(const v2f*)(Whh1 + (n     ) * HID + k0);
    wh1[c][1] = *(const v2f*)(Whh1 + (n + 16) * HID + k0);
    wf[c]     = *(const v2f*)(Wfc  + k0);                    // fc weights, lane's K-half
  }
  const float bias0A = bih0[n]      + bhh0[n];
  const float bias0B = bih0[n + 16] + bhh0[n + 16];
  const float bias1A = bih1[n]      + bhh1[n];
  const float bias1B = bih1[n + 16] + bhh1[n + 16];
  const float wihA   = Wih0[n];        // input weight, column j = n
  const float wihB   = Wih0[n + 16];   // input weight, column j = n+16
  const float bfc0   = bfc[0];

  // ---- Initial hidden state into A-layout registers ----
  v2f a0[8], a1[8];
  #pragma unroll
  for (int c = 0; c < 8; ++c) {
    const int k0 = 4 * c + 2 * hi;
    a0[c] = *(const v2f*)(hid + (b0 + n) * HID + k0);
    a1[c] = *(const v2f*)(hid + (BB + b0 + n) * HID + k0);
  }

  // stage x[.,0]; all lanes participate (halves write identical values)
  xbuf[0][n] = xg[(size_t)(b0 + n) * TT + 0];
  __syncthreads();

  #pragma unroll 1
  for (int t = 0; t < TT; ++t) {
    const int cur = t & 1, nxt = cur ^ 1;

    // branchless prefetch of next timestep's x (clamped; last iter value unused)
    int tn = t + 1; if (tn > TT - 1) tn = TT - 1;
    const float xn = xg[(size_t)(b0 + n) * TT + tn];

    // ---- layer 0: pre = H0 @ Whh0^T + x*Wih0^T + (b_ih0 + b_hh0) ----
    v8f c00, c01;
    #pragma unroll
    for (int r = 0; r < 8; ++r) { c00[r] = bias0A; c01[r] = bias0B; }
    #pragma unroll
    for (int c = 0; c < 8; ++c) {
      c00 = wmma_f32(a0[c], wh0[c][0], c00);
      c01 = wmma_f32(a0[c], wh0[c][1], c01);
    }
    #pragma unroll
    for (int r = 0; r < 8; ++r) {           // D-layout: vgpr r -> batch row m = r + 8*hi
      const float xv = xbuf[cur][r + 8 * hi];
      c00[r] = fast_tanh(c00[r] + xv * wihA);
      c01[r] = fast_tanh(c01[r] + xv * wihB);
    }

    // D-layout -> LDS -> A-layout transpose of new H0
    __syncthreads();                        // prior-iteration hst readers done
    #pragma unroll
    for (int r = 0; r < 8; ++r) {
      hst[r + 8 * hi][n]      = c00[r];
      hst[r + 8 * hi][n + 16] = c01[r];
    }
    __syncthreads();
    #pragma unroll
    for (int c = 0; c < 8; ++c)
      a0[c] = *(const v2f*)&hst[n][4 * c + 2 * hi];

    // ---- layer 1: pre = H0new @ Wih1^T + H1 @ Whh1^T + (b_ih1 + b_hh1) ----
    // Two independent 8-deep WMMA chains per tile (a0-chain seeds bias, a1-chain seeds 0),
    // merged with a vector add: halves the dependent accumulation depth vs one 16-deep chain.
    v8f c10a, c10b, c11a, c11b;
    #pragma unroll
    for (int r = 0; r < 8; ++r) {
      c10a[r] = bias1A; c11a[r] = bias1B;
      c10b[r] = 0.0f;   c11b[r] = 0.0f;
    }
    #pragma unroll
    for (int c = 0; c < 8; ++c) {
      c10a = wmma_f32(a0[c], wi1[c][0], c10a);
      c10b = wmma_f32(a1[c], wh1[c][0], c10b);
      c11a = wmma_f32(a0[c], wi1[c][1], c11a);
      c11b = wmma_f32(a1[c], wh1[c][1], c11b);
    }
    #pragma unroll
    for (int r = 0; r < 8; ++r) {
      c10a[r] = fast_tanh(c10a[r] + c10b[r]);
      c11a[r] = fast_tanh(c11a[r] + c11b[r]);
    }

    // D-layout -> LDS -> A-layout transpose of new H1 (+ stage next x, all lanes)
    __syncthreads();                        // a0 reads of hst done before overwrite
    #pragma unroll
    for (int r = 0; r < 8; ++r) {
      hst[r + 8 * hi][n]      = c10a[r];
      hst[r + 8 * hi][n + 16] = c11a[r];
    }
    xbuf[nxt][n] = xn;                      // duplicate same-value write from both halves
    __syncthreads();

    // read back H1 in A-layout and fuse the fc projection (out = H1 . Wfc + b)
    float acc = 0.0f;
    #pragma unroll
    for (int c = 0; c < 8; ++c) {
      a1[c] = *(const v2f*)&hst[n][4 * c + 2 * hi];
      acc += a1[c].x * wf[c].x + a1[c].y * wf[c].y;   // lane's 16 of 32 K-values
    }
    acc += __shfl_xor(acc, 16, 32);                   // combine the two K-halves
    if (lane < 16) out[(size_t)(b0 + lane) * TT + t] = acc + bfc0;
  }

  // ---- final hidden state: d_out[B*T ..] = [2,B,H] ----
  float* hout = out + (size_t)BB * TT;
  #pragma unroll
  for (int c = 0; c < 8; ++c) {
    const int k0 = 4 * c + 2 * hi;
    *(v2f*)(hout + (b0 + n) * HID + k0)      = a0[c];
    *(v2f*)(hout + (BB + b0 + n) * HID + k0) = a1[c];
  }
}

extern "C" void kernel_launch(void* const* d_in, const int* in_sizes, int n_in,
                              void* d_out, int out_size, void* d_ws, size_t ws_size,
                              hipStream_t stream) {
  const float* x    = (const float*)d_in[0];
  const float* hid  = (const float*)d_in[1];
  const float* Wih0 = (const float*)d_in[2];
  const float* Whh0 = (const float*)d_in[3];
  const float* bih0 = (const float*)d_in[4];
  const float* bhh0 = (const float*)d_in[5];
  const float* Wih1 = (const float*)d_in[6];
  const float* Whh1 = (const float*)d_in[7];
  const float* bih1 = (const float*)d_in[8];
  const float* bhh1 = (const float*)d_in[9];
  const float* Wfc  = (const float*)d_in[10];
  const float* bfc  = (const float*)d_in[11];
  float* out = (float*)d_out;

  dim3 grid(BB / 16), block(32);
  rnn2_wmma_kernel<<<grid, block, 0, stream>>>(x, hid, Wih0, Whh0, bih0, bhh0,
                                               Wih1, Whh1, bih1, bhh1, Wfc, bfc, out);
}